// QuanvolutionFilter_65481071396882
// MI455X (gfx1250) — compile-verified
//
#include <hip/hip_runtime.h>
#include <hip/hip_bf16.h>

typedef __attribute__((ext_vector_type(2))) float v2f;
typedef __attribute__((ext_vector_type(8))) float v8f;

// ---------------------------------------------------------------------------
// Kernel 1: build the fixed 16x16 circuit matrix U from weights[2][4].
// State index i = a*8 + b*4 + c*2 + d (qubit k -> bit (3-k)).
// ---------------------------------------------------------------------------
__global__ void build_u_kernel(const float* __restrict__ w, float* __restrict__ U) {
    if (threadIdx.x != 0) return;
    float M[16][16];
    for (int i = 0; i < 16; ++i)
        for (int j = 0; j < 16; ++j)
            M[i][j] = (i == j) ? 1.0f : 0.0f;

    for (int layer = 0; layer < 2; ++layer) {
        // RY(weights[layer][q]) on each wire q (left-multiply)
        for (int q = 0; q < 4; ++q) {
            float h = 0.5f * w[layer * 4 + q];
            float sg, cg;
            __sincosf(h, &sg, &cg);
            int bit = 8 >> q;
            for (int i = 0; i < 16; ++i) {
                if (i & bit) continue;
                int i1 = i | bit;
                for (int j = 0; j < 16; ++j) {
                    float a0 = M[i][j], a1 = M[i1][j];
                    M[i][j]  = cg * a0 - sg * a1;   // new |0> row
                    M[i1][j] = sg * a0 + cg * a1;   // new |1> row
                }
            }
        }
        // CNOT(0,1): for rows with qubit0==1, swap qubit1 bit
        for (int i = 0; i < 16; ++i) {
            if ((i & 8) && !(i & 4)) {
                int i1 = i | 4;
                for (int j = 0; j < 16; ++j) {
                    float t = M[i][j]; M[i][j] = M[i1][j]; M[i1][j] = t;
                }
            }
        }
        // CNOT(2,3): for rows with qubit2==1, swap qubit3 bit
        for (int i = 0; i < 16; ++i) {
            if ((i & 2) && !(i & 1)) {
                int i1 = i | 1;
                for (int j = 0; j < 16; ++j) {
                    float t = M[i][j]; M[i][j] = M[i1][j]; M[i1][j] = t;
                }
            }
        }
    }
    for (int i = 0; i < 16; ++i)
        for (int j = 0; j < 16; ++j)
            U[i * 16 + j] = M[i][j];
}

// ---------------------------------------------------------------------------
// Kernel 2: per wave, 16 patches -> B(16x16); D = U * B via 4x WMMA F32
// 16x16x4; then z_w = signed sums of D^2 per column.
// Grid geometry fixed by the reference: x is [64,1,256,256].
// ---------------------------------------------------------------------------
__global__ __launch_bounds__(256) void quanv_kernel(const float* __restrict__ x,
                                                    const float* __restrict__ Ug,
                                                    float* __restrict__ out) {
    __shared__ float sU[256];
    const int tid = threadIdx.x;
    sU[tid] = Ug[tid];
    __syncthreads();

    const int lane = tid & 31;
    const int wave = tid >> 5;
    const int L  = lane & 15;   // column (patch within tile) / row of A
    const int hi = lane >> 4;   // lower/upper half of the wave

    // A operand = U, 16x4 chunks. Documented f32 A layout:
    //   v0: K=4kk+0 (lanes 0-15) / K=4kk+2 (lanes 16-31); v1: +1 / +3.
    v2f a[4];
#pragma unroll
    for (int kk = 0; kk < 4; ++kk) {
        a[kk].x = sU[L * 16 + 4 * kk + 2 * hi + 0];
        a[kk].y = sU[L * 16 + 4 * kk + 2 * hi + 1];
    }

    // Patch for this lane's column.
    const long tile = (long)blockIdx.x * 8 + wave;
    const long n    = tile * 16 + L;            // patch id
    const int  b    = (int)(n >> 14);           // 128*128 patches per image
    const int  r    = (int)((n >> 7) & 127);
    const int  c    = (int)(n & 127);
    const float* px = x + (((long)b * 256 + 2 * r) * 256 + 2 * c);
    const float x0 = px[0], x1 = px[1], x2 = px[256], x3 = px[257];

    float s0, c0, s1, c1, s2, c2, s3, c3;
    __sincosf(0.5f * x0, &s0, &c0);
    __sincosf(0.5f * x1, &s1, &c1);
    __sincosf(0.5f * x2, &s2, &c2);
    __sincosf(0.5f * x3, &s3, &c3);

    // B operand = initial states. B[k][col]: k = a*8+b*4+c*2+d.
    // Chunk kk covers k=4kk..4kk+3 => (a,b) = (kk>>1, kk&1), lane half = c, vreg = d.
    const float q2v = hi ? s2 : c2;
    float pa[4];
    pa[0] = c0 * c1; pa[1] = c0 * s1; pa[2] = s0 * c1; pa[3] = s0 * s1;
    v2f bm[4];
#pragma unroll
    for (int kk = 0; kk < 4; ++kk) {
        const float base = pa[kk] * q2v;
        bm[kk].x = base * c3;   // d = 0
        bm[kk].y = base * s3;   // d = 1
    }

    v8f acc = {0.f, 0.f, 0.f, 0.f, 0.f, 0.f, 0.f, 0.f};
#pragma unroll
    for (int kk = 0; kk < 4; ++kk) {
        acc = __builtin_amdgcn_wmma_f32_16x16x4_f32(
            /*neg_a=*/false, a[kk], /*neg_b=*/false, bm[kk],
            /*c_mod=*/(short)0, acc, /*reuse_a=*/false, /*reuse_b=*/false);
    }

    // Lane holds D[m][L] for m = hi*8 + j. Readout: p = D^2, signed sums.
    float p[8];
#pragma unroll
    for (int j = 0; j < 8; ++j) p[j] = acc[j] * acc[j];

    const float t  = (p[0] + p[1] + p[2] + p[3]) + (p[4] + p[5] + p[6] + p[7]);
    const float ub = (p[0] + p[1] + p[2] + p[3]) - (p[4] + p[5] + p[6] + p[7]);
    const float uc = (p[0] + p[1] - p[2] - p[3]) + (p[4] + p[5] - p[6] - p[7]);
    const float ud = (p[0] - p[1] + p[2] - p[3]) + (p[4] - p[5] + p[6] - p[7]);

    const float tp  = __shfl_xor(t,  16, 32);
    const float ubp = __shfl_xor(ub, 16, 32);
    const float ucp = __shfl_xor(uc, 16, 32);
    const float udp = __shfl_xor(ud, 16, 32);

    if (hi == 0) {
        float4 z;
        z.x = t  - tp;    // qubit0: +(a=0 half, ours) - (a=1 half, partner)
        z.y = ub + ubp;   // qubit1
        z.z = uc + ucp;   // qubit2
        z.w = ud + udp;   // qubit3
        reinterpret_cast<float4*>(out)[n] = z;   // out[n*4 + w]
    }
}

extern "C" void kernel_launch(void* const* d_in, const int* in_sizes, int n_in,
                              void* d_out, int out_size, void* d_ws, size_t ws_size,
                              hipStream_t stream) {
    const float* x   = (const float*)d_in[0];   // [64,1,256,256] float32
    const float* wts = (const float*)d_in[1];   // [2,4] float32
    float* U   = (float*)d_ws;                  // 256 floats scratch
    float* out = (float*)d_out;                 // [64, 65536] float32

    build_u_kernel<<<1, 32, 0, stream>>>(wts, U);

    const long npatch = (long)in_sizes[0] / 4;  // 1,048,576
    const int  blocks = (int)(npatch / 128);    // 8 waves/block * 16 patches/wave
    quanv_kernel<<<blocks, 256, 0, stream>>>(x, U, out);
}